// BlurGenerationPair_2138893714019
// MI455X (gfx1250) — compile-verified
//
#include <hip/hip_runtime.h>
#include <hip/hip_bf16.h>

// Problem constants from the reference.
#define B_ 128
#define T_ 512
#define D_ 128

typedef __attribute__((ext_vector_type(2))) float v2f;
typedef __attribute__((ext_vector_type(4))) float v4f;
typedef __attribute__((ext_vector_type(8))) float v8f;

#define KBLK 32
#define LDS_STRIDE (D_ + 4)   // 132 floats -> 528B rows (16B aligned, bank-skewed)

// avged_seq[b] = BlurMat[b]^T (512x512) @ seq[b] (512x128), fp32 via WMMA.
// Workgroup: 4 waves; wave w owns M rows [blockIdx.y*64 + w*16, +16), full N=128.
// K is stepped by KBLK; the seq K-slab is staged in LDS and shared by all waves.
__global__ __launch_bounds__(128) void blur_gemm_kernel(
    const float* __restrict__ seq,      // [B, T, D]
    const float* __restrict__ blurmat,  // [B, T, T]
    float* __restrict__ avged_seq)      // [B, T, D]
{
    __shared__ float ldsB[KBLK * LDS_STRIDE];

    const int b    = blockIdx.x;
    const int tid  = threadIdx.x;
    const int wave = tid >> 5;
    const int lane = tid & 31;
    const int half = lane >> 4;   // which 16-lane half of the wave
    const int l    = lane & 15;
    const int m0   = blockIdx.y * 64 + wave * 16;

    const float* __restrict__ seqb = seq     + (size_t)b * T_ * D_;
    const float* __restrict__ bm   = blurmat + (size_t)b * T_ * T_;

    v8f c[8];
    #pragma unroll
    for (int nt = 0; nt < 8; ++nt) {
        v8f z = {};
        c[nt] = z;
    }

    for (int k0 = 0; k0 < T_; k0 += KBLK) {
        // ---- stage seq rows [k0, k0+KBLK) x [0,128) into LDS (float4, coalesced)
        __syncthreads();   // previous iteration's LDS reads done before overwrite
        const v4f* __restrict__ src = (const v4f*)(seqb + (size_t)k0 * D_);
        #pragma unroll
        for (int it = 0; it < (KBLK * D_ / 4) / 128; ++it) {   // 8 iterations
            int j   = tid + it * 128;   // float4 index within the slab
            int row = j >> 5;           // 32 float4 per row
            int c4  = j & 31;
            *(v4f*)&ldsB[row * LDS_STRIDE + c4 * 4] = src[j];
        }
        __syncthreads();

        // Prefetch next BlurMat K-slab (speculative, RT).
        if (k0 + KBLK < T_) {
            __builtin_prefetch(bm + (size_t)(k0 + KBLK + 2 * half) * T_ + m0 + l, 0, 1);
        }

        // ---- WMMA over this K-slab
        #pragma unroll
        for (int kk = 0; kk < KBLK; kk += 4) {
            const int k = k0 + kk;
            // A fragment (16x4 f32): row M = m0+l, x -> K=k+2*half, y -> K=k+2*half+1.
            // A[i][t] = BlurMat[t*T + i]; lanes 0..15 read consecutive columns -> coalesced.
            const float* __restrict__ ap = bm + (size_t)(k + 2 * half) * T_ + (m0 + l);
            v2f a;
            a.x = ap[0];
            a.y = ap[T_];

            #pragma unroll
            for (int nt = 0; nt < 8; ++nt) {
                // B fragment (4x16 f32) from LDS: N = nt*16 + l, same per-lane K pairing.
                const float* __restrict__ bp =
                    &ldsB[(kk + 2 * half) * LDS_STRIDE + nt * 16 + l];
                v2f bf;
                bf.x = bp[0];
                bf.y = bp[LDS_STRIDE];
                c[nt] = __builtin_amdgcn_wmma_f32_16x16x4_f32(
                    /*neg_a=*/false, a, /*neg_b=*/false, bf,
                    /*c_mod=*/(short)0, c[nt],
                    /*reuse_a=*/false, /*reuse_b=*/false);
            }
        }
    }

    // ---- store C tiles: element r lives at (M = m0+r+8*half, N = nt*16+l)
    float* __restrict__ outb = avged_seq + (size_t)b * T_ * D_;
    #pragma unroll
    for (int nt = 0; nt < 8; ++nt) {
        #pragma unroll
        for (int r = 0; r < 8; ++r) {
            outb[(size_t)(m0 + r + 8 * half) * D_ + nt * 16 + l] = c[nt][r];
        }
    }
}

// R / avged_R ramps + avged_len passthrough (as float into the f32 out buffer).
__global__ __launch_bounds__(256) void ramp_kernel(
    const int* __restrict__ len_seq,      // [B]
    const int* __restrict__ avged_len,    // [B]
    float* __restrict__ R,                // [B, T]
    float* __restrict__ avged_R,          // [B, T]
    float* __restrict__ avged_len_out)    // [B]
{
    int idx = blockIdx.x * blockDim.x + threadIdx.x;   // over B*T
    int b = idx / T_;
    int t = idx - b * T_;

    int L  = len_seq[b];
    int Ld = (L > 0) ? L : 1;
    R[idx] = (t < L) ? (float)(t + 1) / (float)Ld : 0.0f;

    int A  = avged_len[b];
    int Ad = (A > 0) ? A : 1;
    avged_R[idx] = (t < A) ? (float)(t + 1) / (float)Ad : 0.0f;

    if (t == 0) avged_len_out[b] = (float)A;
}

extern "C" void kernel_launch(void* const* d_in, const int* in_sizes, int n_in,
                              void* d_out, int out_size, void* d_ws, size_t ws_size,
                              hipStream_t stream) {
    // Inputs in setup_inputs() order: seq, len_seq, SeqtoBlur, BlurMat, avged_len
    const float* seq       = (const float*)d_in[0];
    const int*   len_seq   = (const int*)  d_in[1];
    const float* SeqtoBlur = (const float*)d_in[2];
    const float* BlurMat   = (const float*)d_in[3];
    const int*   avged_len = (const int*)  d_in[4];

    // Outputs concatenated flat in return order (all viewed as f32 here):
    float* out       = (float*)d_out;
    float* out_s2b   = out;                                   // [B,T,T]
    float* out_avg   = out_s2b  + (size_t)B_ * T_ * T_;       // [B,T,D]
    float* out_R     = out_avg  + (size_t)B_ * T_ * D_;       // [B,T]
    float* out_aR    = out_R    + (size_t)B_ * T_;            // [B,T]
    float* out_alen  = out_aR   + (size_t)B_ * T_;            // [B]

    // 1) SeqtoBlur passthrough: pure 134MB d2d copy (graph-capture safe).
    hipMemcpyAsync(out_s2b, SeqtoBlur, (size_t)B_ * T_ * T_ * sizeof(float),
                   hipMemcpyDeviceToDevice, stream);

    // 2) Batched fp32 WMMA GEMM.
    dim3 grid(B_, T_ / 64);
    blur_gemm_kernel<<<grid, 128, 0, stream>>>(seq, BlurMat, out_avg);

    // 3) Ramps + avged_len.
    ramp_kernel<<<(B_ * T_) / 256, 256, 0, stream>>>(len_seq, avged_len,
                                                     out_R, out_aR, out_alen);
}